// PFGCN_1864015806535
// MI455X (gfx1250) — compile-verified
//
#include <hip/hip_runtime.h>

// Problem constants (match reference)
#define NN 100000   // nodes
#define EE 1600000  // edges per relation
#define RRR 3       // relations
#define FF 128      // feature size
#define HH 64       // hidden dim
#define OO 32       // out dim
#define BB 50000    // batch nodes

typedef float v2f __attribute__((ext_vector_type(2)));
typedef float v8f __attribute__((ext_vector_type(8)));

// ---------------- utility kernels ----------------

__global__ void k_zero(float* __restrict__ p, long n) {
    long i = (long)blockIdx.x * blockDim.x + threadIdx.x;
    if (i < n) p[i] = 0.0f;
}

__global__ void k_degree(const int* __restrict__ dst, float* __restrict__ deg, int e) {
    int i = blockIdx.x * blockDim.x + threadIdx.x;
    if (i < e) atomicAdd(&deg[dst[i]], 1.0f);
}

__global__ void k_dinv(float* __restrict__ d, int n) {
    int i = blockIdx.x * blockDim.x + threadIdx.x;
    if (i < n) d[i] = rsqrtf(fmaxf(d[i], 1.0f));
}

// ---------------- WMMA fp32 GEMM: C[M,NCOL] = A[M,K] @ W[K,NCOL] + bias ----
// One wave32 per 16x16 output tile; K stepped by 4 via V_WMMA_F32_16X16X4_F32.
// 32-bit A 16x4 layout: lanes 0-15 -> M=lane, {K0,K1}; lanes 16-31 -> {K2,K3}.
// B 4x16 mirrored: lanes 0-15 -> N=lane, rows {K0,K1}; lanes 16-31 -> {K2,K3}.
// C/D: VGPR i, lanes 0-15 -> M=i; lanes 16-31 -> M=i+8.

template<int K, int NCOL>
__global__ void k_gemm_wmma(const float* __restrict__ A,
                            const float* __restrict__ W,
                            const float* __restrict__ bias,
                            float* __restrict__ C) {
    const int lane = threadIdx.x & 31;
    const int half = lane >> 4;
    const int l16  = lane & 15;
    const int row0 = blockIdx.x * 16;
    const int col0 = blockIdx.y * 16;

    v8f c = {};
    const float* arow = A + (size_t)(row0 + l16) * K + 2 * half;
    const float* wcol = W + (size_t)(2 * half) * NCOL + (col0 + l16);

    #pragma unroll
    for (int k0 = 0; k0 < K; k0 += 4) {
        v2f a = *(const v2f*)(arow + k0);                 // K = k0+2h, k0+2h+1
        v2f b;
        b.x = wcol[(size_t)k0 * NCOL];                    // K = k0+2h
        b.y = wcol[(size_t)(k0 + 1) * NCOL];              // K = k0+2h+1
        c = __builtin_amdgcn_wmma_f32_16x16x4_f32(
                /*neg_a=*/false, a, /*neg_b=*/false, b,
                /*c_mod=*/(short)0, c, /*reuse_a=*/false, /*reuse_b=*/false);
    }

    float bv = bias[col0 + l16];
    #pragma unroll
    for (int i = 0; i < 8; ++i) {
        int rr = row0 + i + 8 * half;
        C[(size_t)rr * NCOL + col0 + l16] = c[i] + bv;
    }
}

// ---------------- edge aggregation: agg[dst] += h[src] * dinv[src]*dinv[dst] --
// one thread per (edge, 4-channel chunk); float4 gather + 4 f32 atomics (L2-resident)

template<int C>
__global__ void k_edge_agg(const int* __restrict__ src, const int* __restrict__ dst,
                           const float* __restrict__ dinv, const float* __restrict__ h,
                           float* __restrict__ agg, int e) {
    const int CH4 = C / 4;
    long t = (long)blockIdx.x * blockDim.x + threadIdx.x;
    if (t >= (long)e * CH4) return;
    int ei = (int)(t / CH4);
    int ch = (int)(t % CH4) * 4;
    int s = src[ei], d = dst[ei];
    float w = dinv[s] * dinv[d];
    const float4 hv = *(const float4*)(h + (size_t)s * C + ch);
    float* ap = agg + (size_t)d * C + ch;
    atomicAdd(ap + 0, hv.x * w);
    atomicAdd(ap + 1, hv.y * w);
    atomicAdd(ap + 2, hv.z * w);
    atomicAdd(ap + 3, hv.w * w);
}

// ---------------- gamma mix: h = g*agg + (1-g)*h ----------------

__global__ void k_mix(float* __restrict__ h, const float* __restrict__ agg,
                      const float* __restrict__ gp, long n) {
    long i = (long)blockIdx.x * blockDim.x + threadIdx.x;
    if (i < n) {
        float g = gp[0];
        h[i] = g * agg[i] + (1.0f - g) * h[i];
    }
}

// ---------------- batchnorm: per-channel sum / sumsq partials ----------------

template<int C>
__global__ void k_bn_stats(const float* __restrict__ x, float* __restrict__ stats, int n) {
    const int RPB = 256 / C;                 // rows handled per block-iteration
    int c  = threadIdx.x % C;
    int r0 = blockIdx.x * RPB + threadIdx.x / C;
    float s = 0.0f, s2 = 0.0f;
    for (int row = r0; row < n; row += gridDim.x * RPB) {
        float v = x[(size_t)row * C + c];
        s += v; s2 += v * v;
    }
    atomicAdd(&stats[c], s);
    atomicAdd(&stats[C + c], s2);
}

template<int C>
__global__ void k_bn_apply_relu(float* __restrict__ x, const float* __restrict__ stats,
                                const float* __restrict__ scale, const float* __restrict__ bias,
                                int n) {
    long i = (long)blockIdx.x * blockDim.x + threadIdx.x;
    if (i >= (long)n * C) return;
    int c = (int)(i % C);
    float inv_n = 1.0f / (float)n;
    float m  = stats[c] * inv_n;
    float vv = stats[C + c] * inv_n - m * m;
    float rs = rsqrtf(vv + 1e-5f);
    float y = (x[i] - m) * rs * scale[c] + bias[c];
    x[i] = fmaxf(y, 0.0f);
}

// ---------------- gather + log_softmax over O=32 (one wave32 per row) --------

__global__ void k_logsoftmax_gather(const float* __restrict__ x, const int* __restrict__ batch,
                                    float* __restrict__ out, int r, int nb) {
    int wave = threadIdx.x >> 5;
    int lane = threadIdx.x & 31;
    int b = blockIdx.x * (blockDim.x >> 5) + wave;
    if (b >= nb) return;
    int node = batch[b];
    float v = x[(size_t)node * OO + lane];
    float m = v;
    #pragma unroll
    for (int off = 16; off > 0; off >>= 1) m = fmaxf(m, __shfl_xor(m, off, 32));
    float e = expf(v - m);
    float s = e;
    #pragma unroll
    for (int off = 16; off > 0; off >>= 1) s += __shfl_xor(s, off, 32);
    out[(size_t)b * (RRR * OO) + r * OO + lane] = (v - m) - logf(s);
}

// ---------------- host launcher ----------------

extern "C" void kernel_launch(void* const* d_in, const int* in_sizes, int n_in,
                              void* d_out, int out_size, void* d_ws, size_t ws_size,
                              hipStream_t stream) {
    (void)in_sizes; (void)n_in; (void)out_size; (void)ws_size;

    const float* features = (const float*)d_in[0];   // [N,F]
    const int*   eidx     = (const int*)d_in[1];     // [R,2,E]
    const int*   batch    = (const int*)d_in[2];     // [B]
    const float* W1       = (const float*)d_in[3];   // [R,F,H]
    const float* b1       = (const float*)d_in[4];   // [R,H]
    const float* W2       = (const float*)d_in[5];   // [R,H,O]
    const float* b2       = (const float*)d_in[6];   // [R,O]
    const float* gamma1   = (const float*)d_in[7];   // [R]
    const float* gamma2   = (const float*)d_in[8];   // [R]
    const float* bn1s     = (const float*)d_in[9];   // [R,H]
    const float* bn1b     = (const float*)d_in[10];  // [R,H]
    const float* bn2s     = (const float*)d_in[11];  // [R,O]
    const float* bn2b     = (const float*)d_in[12];  // [R,O]
    float* out = (float*)d_out;                      // [B, R*O]

    // workspace carve-out (~77 MB of floats)
    float* ws   = (float*)d_ws;
    float* dinv = ws;  ws += NN;
    float* h1   = ws;  ws += (size_t)NN * HH;
    float* agg1 = ws;  ws += (size_t)NN * HH;
    float* h2   = ws;  ws += (size_t)NN * OO;
    float* agg2 = ws;  ws += (size_t)NN * OO;
    float* stats = ws; // 2*HH floats, reused as 2*OO

    const int TB = 256;
    const long nH = (long)NN * HH, nO = (long)NN * OO;

    for (int r = 0; r < RRR; ++r) {
        const int* src = eidx + (size_t)r * 2 * EE;
        const int* dst = src + EE;

        // symmetric-normalization degrees: dinv = rsqrt(max(indeg,1))
        k_zero<<<(NN + TB - 1) / TB, TB, 0, stream>>>(dinv, NN);
        k_degree<<<(EE + TB - 1) / TB, TB, 0, stream>>>(dst, dinv, EE);
        k_dinv<<<(NN + TB - 1) / TB, TB, 0, stream>>>(dinv, NN);

        // ---- conv1: h1 = features @ W1[r] + b1[r] (WMMA f32) ----
        k_gemm_wmma<FF, HH><<<dim3(NN / 16, HH / 16), 32, 0, stream>>>(
            features, W1 + (size_t)r * FF * HH, b1 + (size_t)r * HH, h1);
        k_zero<<<(int)((nH + TB - 1) / TB), TB, 0, stream>>>(agg1, nH);
        k_edge_agg<HH><<<(int)(((long)EE * (HH / 4) + TB - 1) / TB), TB, 0, stream>>>(
            src, dst, dinv, h1, agg1, EE);
        k_mix<<<(int)((nH + TB - 1) / TB), TB, 0, stream>>>(h1, agg1, gamma1 + r, nH);
        k_zero<<<1, 2 * HH, 0, stream>>>(stats, 2 * HH);
        k_bn_stats<HH><<<512, TB, 0, stream>>>(h1, stats, NN);
        k_bn_apply_relu<HH><<<(int)((nH + TB - 1) / TB), TB, 0, stream>>>(
            h1, stats, bn1s + (size_t)r * HH, bn1b + (size_t)r * HH, NN);

        // ---- conv2: h2 = h1 @ W2[r] + b2[r] (WMMA f32) ----
        k_gemm_wmma<HH, OO><<<dim3(NN / 16, OO / 16), 32, 0, stream>>>(
            h1, W2 + (size_t)r * HH * OO, b2 + (size_t)r * OO, h2);
        k_zero<<<(int)((nO + TB - 1) / TB), TB, 0, stream>>>(agg2, nO);
        k_edge_agg<OO><<<(int)(((long)EE * (OO / 4) + TB - 1) / TB), TB, 0, stream>>>(
            src, dst, dinv, h2, agg2, EE);
        k_mix<<<(int)((nO + TB - 1) / TB), TB, 0, stream>>>(h2, agg2, gamma2 + r, nO);
        k_zero<<<1, 2 * OO, 0, stream>>>(stats, 2 * OO);
        k_bn_stats<OO><<<512, TB, 0, stream>>>(h2, stats, NN);
        k_bn_apply_relu<OO><<<(int)((nO + TB - 1) / TB), TB, 0, stream>>>(
            h2, stats, bn2s + (size_t)r * OO, bn2b + (size_t)r * OO, NN);

        // ---- gather batch rows, log_softmax, write output slice ----
        k_logsoftmax_gather<<<(BB + 7) / 8, TB, 0, stream>>>(h2, batch, out, r, BB);
    }
}